// EF_block_50079318671685
// MI455X (gfx1250) — compile-verified
//
#include <hip/hip_runtime.h>
#include <hip/hip_bf16.h>

// ---------------------------------------------------------------------------
// Problem constants (from reference setup_inputs)
// ---------------------------------------------------------------------------
#define B_   8
#define CIN  256
#define H_   128
#define W_   128
#define HW   (H_ * W_)      // 16384
#define C2   128
#define HU   (2 * H_)       // 256
#define WU   (2 * W_)       // 256

typedef __attribute__((ext_vector_type(2))) float v2f;
typedef __attribute__((ext_vector_type(8))) float v8f;

// ---------------------------------------------------------------------------
// Kernel 1: 1x1 conv (channel GEMV) via V_WMMA_F32_16X16X4_F32.
// One wave = 16 pixels. A[m,k] = x1[b, kc+k, pix_base+m] (fp32, 2 VGPRs/lane),
// B[k,n] = w1[kc+k] broadcast over all n. Accumulate 64 k-chunks of 4.
// A-layout (ISA 7.12.2, 32-bit 16x4): lane<16 -> K=0,1 in v0,v1; lane>=16 -> K=2,3.
// B mirrors the row-striped layout, so both use k0 = kc + 2*(lane>=16).
// D column N=0: lane 0 holds M=0..7 in v[0..7], lane 16 holds M=8..15.
// ---------------------------------------------------------------------------
__global__ __launch_bounds__(256)
void conv1x1_wmma_kernel(const float* __restrict__ x1,
                         const float* __restrict__ w1,
                         const float* __restrict__ b1,
                         float* __restrict__ y)
{
    const int lane   = threadIdx.x & 31;
    const int wave   = (blockIdx.x * blockDim.x + threadIdx.x) >> 5;
    const int pixBase = wave * 16;                 // flat pixel over (b, h*w)
    const int b   = pixBase / HW;                  // HW % 16 == 0 -> no b crossing
    const int hw  = pixBase - b * HW;
    const float* xb = x1 + (size_t)b * CIN * HW + hw;

    const int m     = lane & 15;                   // pixel within tile (M)
    const int kHalf = lane >> 4;                   // 0 -> K{0,1}, 1 -> K{2,3}

    v8f acc = {0.f, 0.f, 0.f, 0.f, 0.f, 0.f, 0.f, 0.f};

    for (int kc = 0; kc < CIN; kc += 4) {
        const int k0 = kc + (kHalf << 1);
        v2f a, w;
        // lanes 0-15 read 16 consecutive floats of channel k0 (64B coalesced),
        // lanes 16-31 the same for channel k0+2.
        a.x = xb[(size_t)(k0 + 0) * HW + m];
        a.y = xb[(size_t)(k0 + 1) * HW + m];
        w.x = w1[k0 + 0];
        w.y = w1[k0 + 1];
        acc = __builtin_amdgcn_wmma_f32_16x16x4_f32(
                  /*neg_a=*/false, a, /*neg_b=*/false, w,
                  /*c_mod=*/(short)0, acc, /*reuse_a=*/false, /*reuse_b=*/false);
    }

    const float bias = b1[0];
    if (lane == 0) {
#pragma unroll
        for (int r = 0; r < 8; ++r) y[pixBase + r] = acc[r] + bias;
    } else if (lane == 16) {
#pragma unroll
        for (int r = 0; r < 8; ++r) y[pixBase + 8 + r] = acc[r] + bias;
    }
}

// ---------------------------------------------------------------------------
// Kernel 2: fused 2x2 stride-2 upsample + 3x3 joint bilateral (reflect pad,
// sigma = 0.8 for both spatial and range) + sigmoid gate + x2 * (1 + gate).
// Block = 256 threads: 4 rows x 64 float4 columns of the 256x256 plane;
// each thread computes 4 gates, then streams all 128 channels of x2 (b128).
// ---------------------------------------------------------------------------
__device__ __forceinline__ float upsample_at(const float* __restrict__ y,
                                             const float* __restrict__ wt,
                                             float bt, int b, int hu, int wu)
{
    const int h = hu >> 1, w = wu >> 1;
    return y[((size_t)b * H_ + h) * W_ + w] * wt[((hu & 1) << 1) | (wu & 1)] + bt;
}

__device__ __forceinline__ int reflect_idx(int t, int n)
{
    t = (t < 0) ? -t : t;
    return (t >= n) ? (2 * n - 2 - t) : t;
}

__global__ __launch_bounds__(256)
void bilateral_gate_kernel(const float* __restrict__ y,
                           const float* __restrict__ x2,
                           const float* __restrict__ wtp,
                           const float* __restrict__ btp,
                           float* __restrict__ out)
{
    // grid: B_ * (HU/4) blocks; block covers rows [huBase, huBase+4)
    const int b      = blockIdx.x >> 6;
    const int huBase = (blockIdx.x & 63) << 2;
    const int hu     = huBase + (threadIdx.x >> 6);
    const int wu4    = (threadIdx.x & 63) << 2;

    float wt[4];
#pragma unroll
    for (int i = 0; i < 4; ++i) wt[i] = wtp[i];
    const float bt = btp[0];

    const float sigma   = 0.15f * 3.0f + 0.35f;          // 0.8
    const float inv2s2  = 1.0f / (2.0f * sigma * sigma); // 0.78125
    // 1-D spatial kernel {e^-inv2s2, 1, e^-inv2s2}; normalization cancels in
    // (sum w*p)/(sum w), so skip it.
    float ks[3];
    ks[0] = __expf(-inv2s2); ks[1] = 1.0f; ks[2] = ks[0];

    float scale[4];
#pragma unroll
    for (int q = 0; q < 4; ++q) {
        const int wu = wu4 + q;
        const float center = upsample_at(y, wt, bt, b, hu, wu);
        float num = 0.0f, den = 0.0f;
#pragma unroll
        for (int di = -1; di <= 1; ++di) {
            const int hr = reflect_idx(hu + di, HU);
#pragma unroll
            for (int dj = -1; dj <= 1; ++dj) {
                const int wr = reflect_idx(wu + dj, WU);
                const float p  = upsample_at(y, wt, bt, b, hr, wr);
                const float d  = p - center;
                const float wk = __expf(-d * d * inv2s2) * ks[di + 1] * ks[dj + 1];
                num += wk * p;
                den += wk;
            }
        }
        const float filtered = num / den;
        const float gate = 1.0f / (1.0f + __expf(-filtered));
        scale[q] = 1.0f + gate;                          // gate*x2 + x2
    }

    // Stream 128 channels; b128 loads/stores, coalesced along wu.
    const size_t planeStride = (size_t)HU * WU;
    size_t off = (((size_t)b * C2) * HU + hu) * WU + wu4;
    for (int c = 0; c < C2; ++c, off += planeStride) {
        if (c + 8 < C2)
            __builtin_prefetch(&x2[off + 8 * planeStride], 0, 0); // global_prefetch_b8
        const float4 v = *reinterpret_cast<const float4*>(&x2[off]);
        float4 r;
        r.x = v.x * scale[0];
        r.y = v.y * scale[1];
        r.z = v.z * scale[2];
        r.w = v.w * scale[3];
        *reinterpret_cast<float4*>(&out[off]) = r;
    }
}

// ---------------------------------------------------------------------------
// Launch. d_in order (setup_inputs): x1, x2, w1, b1, wt, bt. Output fp32.
// d_ws holds y: B_*H_*W_ floats = 512 KB.
// ---------------------------------------------------------------------------
extern "C" void kernel_launch(void* const* d_in, const int* in_sizes, int n_in,
                              void* d_out, int out_size, void* d_ws, size_t ws_size,
                              hipStream_t stream)
{
    const float* x1 = (const float*)d_in[0];
    const float* x2 = (const float*)d_in[1];
    const float* w1 = (const float*)d_in[2];
    const float* b1 = (const float*)d_in[3];
    const float* wt = (const float*)d_in[4];
    const float* bt = (const float*)d_in[5];
    float* out = (float*)d_out;
    float* y   = (float*)d_ws;   // [B_, H_, W_] fp32

    // Kernel 1: 131072 pixels / 16 per wave = 8192 waves; 8 waves per block.
    const int waves  = (B_ * HW) / 16;
    const int blocks1 = waves / 8;
    conv1x1_wmma_kernel<<<blocks1, 256, 0, stream>>>(x1, w1, b1, y);

    // Kernel 2: B_ * HU/4 blocks, 256 threads.
    const int blocks2 = B_ * (HU / 4);
    bilateral_gate_kernel<<<blocks2, 256, 0, stream>>>(y, x2, wt, bt, out);
}